// GCNLayer_48541720379661
// MI455X (gfx1250) — compile-verified
//
#include <hip/hip_runtime.h>
#include <hip/hip_bf16.h>

#define LEAKY_SLOPE 0.5f
#define D 128
#define WAVE 32
#define EDGES_PER_BLOCK 8   // 256 threads / wave32

// ---------------------------------------------------------------------------
// Kernel 1: zero the accumulator (d_out). Must run every call: the harness
// does not re-initialize d_out between graph replays.
// ---------------------------------------------------------------------------
__global__ void gcn_zero_kernel(float4* __restrict__ out, int n4) {
    int i = blockIdx.x * blockDim.x + threadIdx.x;
    if (i < n4) {
        out[i] = make_float4(0.f, 0.f, 0.f, 0.f);
    }
}

// ---------------------------------------------------------------------------
// Kernel 2: edge-parallel scatter-add SpMM.
// One wave32 per edge: lane l handles features [4l, 4l+4).
//  - edge stream (rows/cols/vals) loaded non-temporally: streamed once, keep
//    it out of the 192MB L2 so embeds (51MB) + out (51MB) stay resident.
//  - gather: global_load_b128 per lane -> one coalesced 512B row per wave.
//  - scatter: 4x non-returning global_atomic_add_f32 per lane, resolved in L2.
// ---------------------------------------------------------------------------
__global__ void gcn_scatter_kernel(const int*   __restrict__ rows,
                                   const int*   __restrict__ cols,
                                   const float* __restrict__ vals,
                                   const float* __restrict__ embeds,
                                   float*       __restrict__ out,
                                   int E) {
    const int lane = threadIdx.x & (WAVE - 1);
    const int waveInBlock = threadIdx.x >> 5;
    const long long e = (long long)blockIdx.x * EDGES_PER_BLOCK + waveInBlock;
    if (e >= E) return;

    // Uniform per-edge scalars: all 32 lanes hit the same address -> one
    // request. NT hint keeps the 64MB edge stream from polluting L2.
    const int   r = __builtin_nontemporal_load(rows + e);
    const int   c = __builtin_nontemporal_load(cols + e);
    const float w = __builtin_nontemporal_load(vals + e);

    // Gather 16B per lane from the source row (row lives in L2).
    const float4* __restrict__ src =
        (const float4*)(embeds + (size_t)c * D) + lane;
    float4 v = *src;

    v.x *= w; v.y *= w; v.z *= w; v.w *= w;

    float* __restrict__ dst = out + (size_t)r * D + (size_t)lane * 4;
    atomicAdd(dst + 0, v.x);
    atomicAdd(dst + 1, v.y);
    atomicAdd(dst + 2, v.z);
    atomicAdd(dst + 3, v.w);
}

// ---------------------------------------------------------------------------
// Kernel 3: in-place leaky ReLU (negative_slope = 0.5) on the accumulator.
// Separate launch = device-wide completion of all atomics before reading.
// ---------------------------------------------------------------------------
__global__ void gcn_leaky_kernel(float4* __restrict__ out, int n4) {
    int i = blockIdx.x * blockDim.x + threadIdx.x;
    if (i < n4) {
        float4 v = out[i];
        v.x = (v.x > 0.f) ? v.x : LEAKY_SLOPE * v.x;
        v.y = (v.y > 0.f) ? v.y : LEAKY_SLOPE * v.y;
        v.z = (v.z > 0.f) ? v.z : LEAKY_SLOPE * v.z;
        v.w = (v.w > 0.f) ? v.w : LEAKY_SLOPE * v.w;
        out[i] = v;
    }
}

extern "C" void kernel_launch(void* const* d_in, const int* in_sizes, int n_in,
                              void* d_out, int out_size, void* d_ws, size_t ws_size,
                              hipStream_t stream) {
    // Inputs per setup_inputs() order:
    //   d_in[0]: edge_index  int   [2, E] flat  (rows = [0:E), cols = [E:2E))
    //   d_in[1]: edge_vals   float [E]
    //   d_in[2]: embeds      float [N, 128]
    const int*   edge_index = (const int*)d_in[0];
    const float* edge_vals  = (const float*)d_in[1];
    const float* embeds     = (const float*)d_in[2];
    float*       out        = (float*)d_out;

    const int E    = in_sizes[1];        // edge count
    const int*   rows = edge_index;
    const int*   cols = edge_index + E;

    const int n4 = out_size / 4;         // N*D/4 float4 elements

    // 1) zero accumulator
    {
        int threads = 256;
        int blocks  = (n4 + threads - 1) / threads;
        gcn_zero_kernel<<<blocks, threads, 0, stream>>>((float4*)out, n4);
    }

    // 2) scatter-add SpMM: one wave32 per edge, 8 edges per 256-thread block
    {
        int threads = 256;
        int blocks  = (E + EDGES_PER_BLOCK - 1) / EDGES_PER_BLOCK;
        gcn_scatter_kernel<<<blocks, threads, 0, stream>>>(
            rows, cols, edge_vals, embeds, out, E);
    }

    // 3) leaky ReLU in place
    {
        int threads = 256;
        int blocks  = (n4 + threads - 1) / threads;
        gcn_leaky_kernel<<<blocks, threads, 0, stream>>>((float4*)out, n4);
    }
}